// DifferentiableRobotModel_61667140436131
// MI455X (gfx1250) — compile-verified
//
#include <hip/hip_runtime.h>
#include <math.h>

#define JN 32
#define GRAV 9.81f
#define BLOCK 256

#if defined(__AMDGCN__) && __has_builtin(__builtin_amdgcn_global_load_async_to_lds_b128) && __has_builtin(__builtin_amdgcn_s_wait_asynccnt)
#define USE_ASYNC_LDS 1
typedef int v4i __attribute__((ext_vector_type(4)));
typedef __attribute__((address_space(1))) v4i GV4;   // global int4
typedef __attribute__((address_space(3))) v4i LV4;   // LDS int4
#else
#define USE_ASYNC_LDS 0
#endif

// Inverse dynamics for a serial chain of N=32 revolute-z joints.
// One thread per batch element. Forward pass runs the kinematic scan; the
// backward (force) pass re-derives previous joint states by inverting the
// forward recurrence (rotations are orthonormal), so no per-joint state is
// checkpointed anywhere -> pure register machine + tiny LDS param table.
__global__ __launch_bounds__(BLOCK) void rnea_kernel(
    const float* __restrict__ q, const float* __restrict__ qd,
    const float* __restrict__ qdd, const float* __restrict__ trans,
    const float* __restrict__ mass, const float* __restrict__ com,
    const float* __restrict__ inertia, const float* __restrict__ damping,
    float* __restrict__ out, int Btot) {
  // ---- LDS: raw staged params + derived per-joint constants -------------
  __shared__ __attribute__((aligned(16)))
      float s_stage[544];         // [0:96) trans, [96:128) mass, [128:224) com,
                                  // [224:512) inertia, [512:544) damping
  __shared__ float s_p[3 * JN];   // joint translation p
  __shared__ float s_m[JN];       // mass
  __shared__ float s_mc[3 * JN];  // mass * com
  __shared__ float s_Ib[9 * JN];  // inertia + m*((c.c)I - c c^T)
  __shared__ float s_dmp[JN];     // damping

  const int tid = threadIdx.x;

#if USE_ASYNC_LDS
  // Async global->LDS copy (gfx1250 GLOBAL_LOAD_ASYNC_TO_LDS_B128, ASYNCcnt).
  if (tid < 24) __builtin_amdgcn_global_load_async_to_lds_b128(
      (GV4*)(trans + tid * 4), (LV4*)&s_stage[0 + tid * 4], 0, 0);
  if (tid < 8) __builtin_amdgcn_global_load_async_to_lds_b128(
      (GV4*)(mass + tid * 4), (LV4*)&s_stage[96 + tid * 4], 0, 0);
  if (tid < 24) __builtin_amdgcn_global_load_async_to_lds_b128(
      (GV4*)(com + tid * 4), (LV4*)&s_stage[128 + tid * 4], 0, 0);
  if (tid < 72) __builtin_amdgcn_global_load_async_to_lds_b128(
      (GV4*)(inertia + tid * 4), (LV4*)&s_stage[224 + tid * 4], 0, 0);
  if (tid < 8) __builtin_amdgcn_global_load_async_to_lds_b128(
      (GV4*)(damping + tid * 4), (LV4*)&s_stage[512 + tid * 4], 0, 0);
  __builtin_amdgcn_s_wait_asynccnt(0);
#else
  for (int i = tid; i < 96; i += BLOCK) s_stage[i] = trans[i];
  for (int i = tid; i < 32; i += BLOCK) s_stage[96 + i] = mass[i];
  for (int i = tid; i < 96; i += BLOCK) s_stage[128 + i] = com[i];
  for (int i = tid; i < 288; i += BLOCK) s_stage[224 + i] = inertia[i];
  for (int i = tid; i < 32; i += BLOCK) s_stage[512 + i] = damping[i];
#endif
  __syncthreads();

  // ---- derive per-joint constants (threads 0..31) -----------------------
  if (tid < JN) {
    const int j = tid;
    const float m = s_stage[96 + j];
    const float cx = s_stage[128 + 3 * j + 0];
    const float cy = s_stage[128 + 3 * j + 1];
    const float cz = s_stage[128 + 3 * j + 2];
    s_p[3 * j + 0] = s_stage[3 * j + 0];
    s_p[3 * j + 1] = s_stage[3 * j + 1];
    s_p[3 * j + 2] = s_stage[3 * j + 2];
    s_m[j] = m;
    s_dmp[j] = s_stage[512 + j];
    s_mc[3 * j + 0] = m * cx;
    s_mc[3 * j + 1] = m * cy;
    s_mc[3 * j + 2] = m * cz;
    const float c2 = cx * cx + cy * cy + cz * cz;
    const int ib = 224 + 9 * j;
    s_Ib[9 * j + 0] = s_stage[ib + 0] + m * (c2 - cx * cx);
    s_Ib[9 * j + 1] = s_stage[ib + 1] - m * cx * cy;
    s_Ib[9 * j + 2] = s_stage[ib + 2] - m * cx * cz;
    s_Ib[9 * j + 3] = s_stage[ib + 3] - m * cy * cx;
    s_Ib[9 * j + 4] = s_stage[ib + 4] + m * (c2 - cy * cy);
    s_Ib[9 * j + 5] = s_stage[ib + 5] - m * cy * cz;
    s_Ib[9 * j + 6] = s_stage[ib + 6] - m * cz * cx;
    s_Ib[9 * j + 7] = s_stage[ib + 7] - m * cz * cy;
    s_Ib[9 * j + 8] = s_stage[ib + 8] + m * (c2 - cz * cz);
  }
  __syncthreads();

  const int b = blockIdx.x * BLOCK + tid;
  if (b >= Btot) return;

  // ---- load this batch row (contiguous 128B per lane, b128 loads) -------
  float qa[JN], qda[JN], qdda[JN];
  {
    const float4* q4 = (const float4*)(q + (size_t)b * JN);
    const float4* d4 = (const float4*)(qd + (size_t)b * JN);
    const float4* a4 = (const float4*)(qdd + (size_t)b * JN);
#pragma unroll
    for (int k = 0; k < JN / 4; ++k) {
      float4 t = q4[k];
      qa[4 * k] = t.x; qa[4 * k + 1] = t.y; qa[4 * k + 2] = t.z; qa[4 * k + 3] = t.w;
      t = d4[k];
      qda[4 * k] = t.x; qda[4 * k + 1] = t.y; qda[4 * k + 2] = t.z; qda[4 * k + 3] = t.w;
      t = a4[k];
      qdda[4 * k] = t.x; qdda[4 * k + 1] = t.y; qdda[4 * k + 2] = t.z; qdda[4 * k + 3] = t.w;
    }
  }

  // ---- forward kinematic scan ------------------------------------------
  float vlx = 0.f, vly = 0.f, vlz = 0.f;          // v_lin
  float wx = 0.f, wy = 0.f, wz = 0.f;             // v_ang
  float alx = 0.f, aly = 0.f, alz = GRAV;         // a_lin (gravity trick)
  float aax = 0.f, aay = 0.f, aaz = 0.f;          // a_ang
#pragma unroll
  for (int j = 0; j < JN; ++j) {
    float s, c;
    __sincosf(qa[j], &s, &c);
    const float px = s_p[3 * j], py = s_p[3 * j + 1], pz = s_p[3 * j + 2];
    const float tx = -(c * px + s * py);
    const float ty = -(c * py - s * px);
    const float tz = -pz;
    const float wvx = c * wx + s * wy;            // R^T v_ang
    const float wvy = c * wy - s * wx;
    const float wvz = wz;
    const float rvx = c * vlx + s * vly;          // R^T v_lin
    const float rvy = c * vly - s * vlx;
    const float rvz = vlz;
    const float lvx = ty * wvz - tz * wvy + rvx;  // t x wv + R^T v_lin
    const float lvy = tz * wvx - tx * wvz + rvy;
    const float lvz = tx * wvy - ty * wvx + rvz;
    const float wax = c * aax + s * aay;          // R^T a_ang
    const float way = c * aay - s * aax;
    const float waz = aaz;
    const float rax = c * alx + s * aly;          // R^T a_lin
    const float ray = c * aly - s * alx;
    const float raz = alz;
    const float lax = ty * waz - tz * way + rax;
    const float lay = tz * wax - tx * waz + ray;
    const float laz = tx * way - ty * wax + raz;
    const float qdj = qda[j], qddj = qdda[j];
    wx = wvx; wy = wvy; wz = wvz + qdj;           // v_ang'
    vlx = lvx; vly = lvy; vlz = lvz;              // v_lin'
    aax = wax + wy * qdj;                         // a_ang' (+ v_ang' x jv)
    aay = way - wx * qdj;
    aaz = waz + qddj;
    alx = lax + vly * qdj;                        // a_lin' (+ v_lin' x jv)
    aly = lay - vlx * qdj;
    alz = laz;
  }

  // ---- backward force scan, inverting the forward recurrence -----------
  float glx = 0.f, gly = 0.f, glz = 0.f;
  float gax = 0.f, gay = 0.f, gaz = 0.f;
  float tau[JN];
#pragma unroll
  for (int j = JN - 1; j >= 0; --j) {
    const float m = s_m[j];
    const float mcx = s_mc[3 * j], mcy = s_mc[3 * j + 1], mcz = s_mc[3 * j + 2];
    const float I0 = s_Ib[9 * j + 0], I1 = s_Ib[9 * j + 1], I2 = s_Ib[9 * j + 2];
    const float I3 = s_Ib[9 * j + 3], I4 = s_Ib[9 * j + 4], I5 = s_Ib[9 * j + 5];
    const float I6 = s_Ib[9 * j + 6], I7 = s_Ib[9 * j + 7], I8 = s_Ib[9 * j + 8];
    const float falx = m * alx + (mcy * aaz - mcz * aay);
    const float faly = m * aly + (mcz * aax - mcx * aaz);
    const float falz = m * alz + (mcx * aay - mcy * aax);
    const float faax = I0 * aax + I1 * aay + I2 * aaz + (mcy * alz - mcz * aly);
    const float faay = I3 * aax + I4 * aay + I5 * aaz + (mcz * alx - mcx * alz);
    const float faaz = I6 * aax + I7 * aay + I8 * aaz + (mcx * aly - mcy * alx);
    const float fvlx = m * vlx + (mcy * wz - mcz * wy);
    const float fvly = m * vly + (mcz * wx - mcx * wz);
    const float fvlz = m * vlz + (mcx * wy - mcy * wx);
    const float fvax = I0 * wx + I1 * wy + I2 * wz + (mcy * vlz - mcz * vly);
    const float fvay = I3 * wx + I4 * wy + I5 * wz + (mcz * vlx - mcx * vlz);
    const float fvaz = I6 * wx + I7 * wy + I8 * wz + (mcx * vly - mcy * vlx);
    const float cfax = (wy * fvaz - wz * fvay) + (vly * fvlz - vlz * fvly);
    const float cfay = (wz * fvax - wx * fvaz) + (vlz * fvlx - vlx * fvlz);
    const float cfaz = (wx * fvay - wy * fvax) + (vlx * fvly - vly * fvlx);
    const float cflx = wy * fvlz - wz * fvly;
    const float cfly = wz * fvlx - wx * fvlz;
    const float cflz = wx * fvly - wy * fvlx;
    const float flx = falx + cflx + glx;
    const float fly = faly + cfly + gly;
    const float flz = falz + cflz + glz;
    const float fax = faax + cfax + gax;
    const float fay = faay + cfay + gay;
    const float faz = faaz + cfaz + gaz;
    tau[j] = faz + s_dmp[j] * qda[j];

    float s, c;
    __sincosf(qa[j], &s, &c);
    const float px = s_p[3 * j], py = s_p[3 * j + 1], pz = s_p[3 * j + 2];
    const float Rfx = c * flx - s * fly;          // R f_lin
    const float Rfy = s * flx + c * fly;
    const float Rfz = flz;
    const float Rax = c * fax - s * fay;          // R f_ang
    const float Ray = s * fax + c * fay;
    const float Raz = faz;
    glx = Rfx; gly = Rfy; glz = Rfz;
    gax = (py * Rfz - pz * Rfy) + Rax;            // p x Rf + R f_ang
    gay = (pz * Rfx - px * Rfz) + Ray;
    gaz = (px * Rfy - py * Rfx) + Raz;

    if (j > 0) {  // invert forward step j -> state_{j-1}
      const float qdj = qda[j], qddj = qdda[j];
      const float wvx = wx, wvy = wy, wvz = wz - qdj;       // R^T w_prev
      const float wax = aax - wy * qdj;                     // R^T-frame ang acc
      const float way = aay + wx * qdj;
      const float waz = aaz - qddj;
      const float lax = alx - vly * qdj;                    // la
      const float lay = aly + vlx * qdj;
      const float laz = alz;
      const float tx = -(c * px + s * py);
      const float ty = -(c * py - s * px);
      const float tz = -pz;
      const float ux = vlx - (ty * wvz - tz * wvy);         // R^T v_prev
      const float uy = vly - (tz * wvx - tx * wvz);
      const float uz = vlz - (tx * wvy - ty * wvx);
      const float bx = lax - (ty * waz - tz * way);         // R^T a_prev
      const float by = lay - (tz * wax - tx * waz);
      const float bz = laz - (tx * way - ty * wax);
      vlx = c * ux - s * uy; vly = s * ux + c * uy; vlz = uz;
      alx = c * bx - s * by; aly = s * bx + c * by; alz = bz;
      wx = c * wvx - s * wvy; wy = s * wvx + c * wvy; wz = wvz;
      aax = c * wax - s * way; aay = s * wax + c * way; aaz = waz;
    }
  }

  // ---- coalesced 128-bit stores ----------------------------------------
  float4* o4 = (float4*)(out + (size_t)b * JN);
#pragma unroll
  for (int k = 0; k < JN / 4; ++k)
    o4[k] = make_float4(tau[4 * k], tau[4 * k + 1], tau[4 * k + 2], tau[4 * k + 3]);
}

extern "C" void kernel_launch(void* const* d_in, const int* in_sizes, int n_in,
                              void* d_out, int out_size, void* d_ws, size_t ws_size,
                              hipStream_t stream) {
  (void)n_in; (void)d_ws; (void)ws_size; (void)out_size;
  const float* q = (const float*)d_in[0];
  const float* qd = (const float*)d_in[1];
  const float* qdd = (const float*)d_in[2];
  const float* trans = (const float*)d_in[3];
  const float* mass = (const float*)d_in[4];
  const float* com = (const float*)d_in[5];
  const float* inertia = (const float*)d_in[6];
  const float* damping = (const float*)d_in[7];
  const int Btot = in_sizes[0] / JN;
  const int grid = (Btot + BLOCK - 1) / BLOCK;
  rnea_kernel<<<grid, BLOCK, 0, stream>>>(q, qd, qdd, trans, mass, com,
                                          inertia, damping, (float*)d_out, Btot);
}